// LuongAttentionDecoder_37529424232686
// MI455X (gfx1250) — compile-verified
//
#include <hip/hip_runtime.h>
#include <hip/hip_bf16.h>
#include <math.h>

// ---------------------------------------------------------------------------
// LuongAttentionDecoder single step, MI455X (gfx1250, wave32, WMMA).
// H=1024, V=32000, B=64, S=512, E=2048.
// Entirely HBM-bandwidth bound (~1 GB streamed fp32) -> use native fp32 WMMA
// (V_WMMA_F32_16X16X4_F32) so math never becomes the bottleneck and no
// precision is lost. Attention is refactored: q = h_new @ attn_w, then
// scores[b,s] = q[b] . enc[s,b]  (attn_b cancels under softmax).
// ---------------------------------------------------------------------------

typedef __attribute__((ext_vector_type(2))) float v2f;
typedef __attribute__((ext_vector_type(8))) float v8f;

__device__ __forceinline__ v8f wmma_f32_k4(v2f a, v2f b, v8f c) {
  // D = A(16x4) x B(4x16) + C(16x16), all fp32.
  return __builtin_amdgcn_wmma_f32_16x16x4_f32(
      /*neg_a=*/false, a, /*neg_b=*/false, b,
      /*c_mod=*/(short)0, c, /*reuse_a=*/false, /*reuse_b=*/false);
}

// ---------------------------------------------------------------------------
// Generic D = act( A@W^T (+ A2@W2^T) + bias (+bias2) ), A:[64,K], W:[N,K] rowmajor.
// One wave per 16x16 tile; block = 8 waves covering 128 output columns.
// grid = (N/128, 64/16).
// ---------------------------------------------------------------------------
__global__ __launch_bounds__(256) void wmma_gemm_wt(
    const float* __restrict__ A,  const float* __restrict__ W,
    const float* __restrict__ A2, const float* __restrict__ W2,
    const float* __restrict__ bias, const float* __restrict__ bias2,
    float* __restrict__ D, int K, int N, int act)
{
  const int lane = threadIdx.x & 31;
  const int wave = threadIdx.x >> 5;
  const int n0   = (blockIdx.x * 8 + wave) * 16;
  const int m0   = blockIdx.y * 16;
  const int half = lane >> 4;        // 0: lanes 0-15, 1: lanes 16-31
  const int lid  = lane & 15;
  const int koff = half * 2;         // ISA 32-bit A/B fragment K sub-offset
  const int arow = m0 + lid;
  const int ncol = n0 + lid;

  const float* a_ptr = A + (size_t)arow * K;
  const float* w_ptr = W + (size_t)ncol * K;

  v8f c = {0.f, 0.f, 0.f, 0.f, 0.f, 0.f, 0.f, 0.f};
  for (int k = 0; k < K; k += 4) {
    v2f a = *(const v2f*)(a_ptr + k + koff);
    v2f b = *(const v2f*)(w_ptr + k + koff);
    c = wmma_f32_k4(a, b, c);
  }
  if (A2) {  // fused second GEMM into same accumulator (LSTM x@Wih^T + h@Whh^T)
    const float* a2 = A2 + (size_t)arow * K;
    const float* w2 = W2 + (size_t)ncol * K;
    for (int k = 0; k < K; k += 4) {
      v2f a = *(const v2f*)(a2 + k + koff);
      v2f b = *(const v2f*)(w2 + k + koff);
      c = wmma_f32_k4(a, b, c);
    }
  }

  float bv = bias ? bias[ncol] : 0.f;
  if (bias2) bv += bias2[ncol];

  #pragma unroll
  for (int r = 0; r < 8; ++r) {      // C/D layout: vgpr r -> row r (+8 for hi half)
    float v = c[r] + bv;
    if (act == 1) v = tanhf(v);
    D[(size_t)(m0 + r + 8 * half) * N + ncol] = v;
  }
}

// ---------------------------------------------------------------------------
// D = A @ W, A:[64,K], W:[K,N] row-major (used for q = h_new @ attn_w).
// ---------------------------------------------------------------------------
__global__ __launch_bounds__(256) void wmma_gemm_kn(
    const float* __restrict__ A, const float* __restrict__ W,
    float* __restrict__ D, int K, int N)
{
  const int lane = threadIdx.x & 31;
  const int wave = threadIdx.x >> 5;
  const int n0   = (blockIdx.x * 8 + wave) * 16;
  const int m0   = blockIdx.y * 16;
  const int half = lane >> 4;
  const int lid  = lane & 15;
  const int koff = half * 2;
  const int arow = m0 + lid;
  const int ncol = n0 + lid;

  const float* a_ptr = A + (size_t)arow * K;

  v8f c = {0.f, 0.f, 0.f, 0.f, 0.f, 0.f, 0.f, 0.f};
  for (int k = 0; k < K; k += 4) {
    const int kr = k + koff;
    v2f a = *(const v2f*)(a_ptr + kr);
    v2f b;
    b.x = W[(size_t)kr * N + ncol];
    b.y = W[(size_t)(kr + 1) * N + ncol];
    c = wmma_f32_k4(a, b, c);
  }
  #pragma unroll
  for (int r = 0; r < 8; ++r)
    D[(size_t)(m0 + r + 8 * half) * N + ncol] = c[r];
}

// x = emb[input_seq]  (64 x 1024)
__global__ void gather_embed(const int* __restrict__ seq,
                             const float* __restrict__ emb,
                             float* __restrict__ x)
{
  int idx = blockIdx.x * blockDim.x + threadIdx.x;   // 65536
  int b = idx >> 10, k = idx & 1023;
  x[idx] = emb[(size_t)seq[b] * 1024 + k];
}

// LSTM pointwise (PyTorch gate order i,f,g,o)
__global__ void lstm_pointwise(const float* __restrict__ gates,
                               const float* __restrict__ c0,
                               float* __restrict__ h_new,
                               float* __restrict__ c_new)
{
  int idx = blockIdx.x * blockDim.x + threadIdx.x;   // 65536
  int b = idx >> 10, hh = idx & 1023;
  const float* g = gates + (size_t)b * 4096;
  float ig = g[hh], fg = g[1024 + hh], gg = g[2048 + hh], og = g[3072 + hh];
  ig = 1.f / (1.f + __expf(-ig));
  fg = 1.f / (1.f + __expf(-fg));
  gg = tanhf(gg);
  og = 1.f / (1.f + __expf(-og));
  float cn = fg * c0[idx] + ig * gg;
  c_new[idx] = cn;
  h_new[idx] = og * tanhf(cn);
}

// scores[b,s] = q[b,:] . enc[s,b,:]   -- one wave per (b,s), float4 + shuffle reduce
__global__ __launch_bounds__(256) void scores_kernel(
    const float* __restrict__ q, const float* __restrict__ enc,
    float* __restrict__ scores)
{
  const int lane = threadIdx.x & 31;
  const int wave = threadIdx.x >> 5;
  const int s = blockIdx.x * 8 + wave;   // 0..511
  const int b = blockIdx.y;              // 0..63
  const float4* qb = (const float4*)(q + (size_t)b * 2048);
  const float4* er = (const float4*)(enc + ((size_t)s * 64 + b) * 2048);
  float acc = 0.f;
  #pragma unroll
  for (int i = lane; i < 512; i += 32) {
    float4 qa = qb[i], ea = er[i];
    acc += qa.x * ea.x + qa.y * ea.y + qa.z * ea.z + qa.w * ea.w;
  }
  #pragma unroll
  for (int off = 16; off > 0; off >>= 1) acc += __shfl_xor(acc, off, 32);
  if (lane == 0) scores[(size_t)b * 512 + s] = acc;
}

// Row softmax over S=512; one wave per b.
__global__ void softmax_kernel(const float* __restrict__ scores,
                               float* __restrict__ aw)
{
  const int b = blockIdx.x;
  const int lane = threadIdx.x;  // blockDim = 32
  float v[16];
  float m = -3.402823466e38f;
  #pragma unroll
  for (int i = 0; i < 16; ++i) {
    v[i] = scores[(size_t)b * 512 + lane + i * 32];
    m = fmaxf(m, v[i]);
  }
  #pragma unroll
  for (int off = 16; off > 0; off >>= 1) m = fmaxf(m, __shfl_xor(m, off, 32));
  float sum = 0.f;
  #pragma unroll
  for (int i = 0; i < 16; ++i) { v[i] = __expf(v[i] - m); sum += v[i]; }
  #pragma unroll
  for (int off = 16; off > 0; off >>= 1) sum += __shfl_xor(sum, off, 32);
  float inv = 1.f / sum;
  #pragma unroll
  for (int i = 0; i < 16; ++i)
    aw[(size_t)b * 512 + lane + i * 32] = v[i] * inv;
}

// context[b,:] = sum_s aw[b,s] * enc[s,b,:]  -- one block per b, coalesced rows
__global__ __launch_bounds__(256) void context_kernel(
    const float* __restrict__ aw, const float* __restrict__ enc,
    float* __restrict__ ctx)
{
  const int b = blockIdx.x;
  const int t = threadIdx.x;             // 0..255 -> 8 e-values each
  float acc[8] = {0.f, 0.f, 0.f, 0.f, 0.f, 0.f, 0.f, 0.f};
  const float* wrow = aw + (size_t)b * 512;
  for (int s = 0; s < 512; ++s) {
    float w = wrow[s];
    const float* er = enc + ((size_t)s * 64 + b) * 2048 + t;
    #pragma unroll
    for (int j = 0; j < 8; ++j) acc[j] += w * er[j * 256];
  }
  #pragma unroll
  for (int j = 0; j < 8; ++j) ctx[(size_t)b * 2048 + t + j * 256] = acc[j];
}

// concat_input = [h_new | context]  (64 x 3072)
__global__ void concat_kernel(const float* __restrict__ h,
                              const float* __restrict__ ctx,
                              float* __restrict__ ci)
{
  int idx = blockIdx.x * blockDim.x + threadIdx.x;   // 196608
  int b = idx / 3072, k = idx - b * 3072;
  ci[idx] = (k < 1024) ? h[(size_t)b * 1024 + k]
                       : ctx[(size_t)b * 2048 + (k - 1024)];
}

extern "C" void kernel_launch(void* const* d_in, const int* in_sizes, int n_in,
                              void* d_out, int out_size, void* d_ws, size_t ws_size,
                              hipStream_t stream) {
  (void)in_sizes; (void)n_in; (void)out_size; (void)ws_size;

  const int*   seq      = (const int*)  d_in[0];
  const float* h0       = (const float*)d_in[1];
  const float* c0       = (const float*)d_in[2];
  const float* enc      = (const float*)d_in[3];
  const float* emb      = (const float*)d_in[4];
  const float* w_ih     = (const float*)d_in[5];
  const float* w_hh     = (const float*)d_in[6];
  const float* b_ih     = (const float*)d_in[7];
  const float* b_hh     = (const float*)d_in[8];
  const float* attn_w   = (const float*)d_in[9];
  // d_in[10] = attn_b: provably cancels (per-b constant shift under softmax).
  const float* concat_w = (const float*)d_in[11];
  const float* concat_b = (const float*)d_in[12];
  const float* out_w    = (const float*)d_in[13];
  const float* out_b    = (const float*)d_in[14];

  float* out   = (float*)d_out;
  float* h_new = out + 2048000;          // [1,64,1024]
  float* c_new = out + 2113536;          // [1,64,1024]
  float* aw    = out + 2179072;          // [64,1,512]

  float* ws     = (float*)d_ws;          // ~4.1 MB used
  float* x      = ws;                    //  65536  (64x1024)
  float* gates  = ws + 65536;            // 262144  (64x4096)
  float* q      = ws + 327680;           // 131072  (64x2048)
  float* scores = ws + 458752;           //  32768  (64x512)
  float* ctx    = ws + 491520;           // 131072  (64x2048)
  float* cin    = ws + 622592;           // 196608  (64x3072)
  float* cc     = ws + 819200;           // 196608  (64x3072)

  // 1) embedding gather
  gather_embed<<<256, 256, 0, stream>>>(seq, emb, x);
  // 2) gates = x@w_ih^T + h0@w_hh^T + b_ih + b_hh   (fused double GEMM)
  wmma_gemm_wt<<<dim3(32, 4), 256, 0, stream>>>(x, w_ih, h0, w_hh, b_ih, b_hh,
                                                gates, 1024, 4096, 0);
  // 3) LSTM pointwise -> h_new, c_new (direct to d_out slots)
  lstm_pointwise<<<256, 256, 0, stream>>>(gates, c0, h_new, c_new);
  // 4) q = h_new @ attn_w   ([64,1024] x [1024,2048])
  wmma_gemm_kn<<<dim3(16, 4), 256, 0, stream>>>(h_new, attn_w, q, 1024, 2048);
  // 5) scores (bandwidth pass #1 over encoder_outputs)
  scores_kernel<<<dim3(64, 64), 256, 0, stream>>>(q, enc, scores);
  // 6) softmax -> attn_weights (d_out slot)
  softmax_kernel<<<64, 32, 0, stream>>>(scores, aw);
  // 7) context (bandwidth pass #2 over encoder_outputs)
  context_kernel<<<64, 256, 0, stream>>>(aw, enc, ctx);
  // 8) concat, tanh GEMM, vocab GEMM
  concat_kernel<<<768, 256, 0, stream>>>(h_new, ctx, cin);
  wmma_gemm_wt<<<dim3(24, 4), 256, 0, stream>>>(cin, concat_w, nullptr, nullptr,
                                                concat_b, nullptr, cc, 3072, 3072, 1);
  wmma_gemm_wt<<<dim3(250, 4), 256, 0, stream>>>(cc, out_w, nullptr, nullptr,
                                                 out_b, nullptr, out, 3072, 32000, 0);
}